// GraphSAGEModel_68427418960017
// MI455X (gfx1250) — compile-verified
//
#include <hip/hip_runtime.h>
#include <hip/hip_bf16.h>
#include <stdint.h>

typedef __attribute__((ext_vector_type(2))) float v2f;
typedef __attribute__((ext_vector_type(8))) float v8f;

#define NNODES 100000
#define NEDGES 3200000LL

// ---------------------------------------------------------------------------
// Fused dual GEMM: Y = X @ Wl, R = X @ Wr  (X: [N,din], W: [din,dout])
// One wave32 per 16x16 output tile; K-loop in steps of 4 with
// V_WMMA_F32_16X16X4_F32. Weights staged once per block into LDS (<=32KB of
// the 320KB/WGP) so B-fragments come from ds_load instead of hammering L2.
// ISA layouts (CDNA5 ISA §7.12.2):
//   A 16x4 f32: lane = khalf*16 + m ; VGPR0=K(2*khalf), VGPR1=K(2*khalf+1)
//   B 4x16 f32: lane = khalf*16 + n ; VGPR0=K(2*khalf), VGPR1=K(2*khalf+1)
//   C/D 16x16 f32: VGPR i -> (M=i, N=lane) lanes 0-15, (M=8+i) lanes 16-31
// ---------------------------------------------------------------------------
__global__ void sage_dual_gemm(const float* __restrict__ X,
                               const float* __restrict__ Wl,
                               const float* __restrict__ Wr,
                               float* __restrict__ Y,
                               float* __restrict__ R,
                               int n, int din, int dout) {
  extern __shared__ float lds[];            // 2 * din * dout floats
  float* sWl = lds;
  float* sWr = lds + din * dout;

  // Cooperative weight staging (whole block, before any divergence).
  const int wElems = din * dout;
  for (int i = threadIdx.x; i < wElems; i += blockDim.x) {
    sWl[i] = Wl[i];
    sWr[i] = Wr[i];
  }
  __syncthreads();

  const int wavesPerBlock = blockDim.x >> 5;
  const int waveId = blockIdx.x * wavesPerBlock + (threadIdx.x >> 5);
  const int lane   = threadIdx.x & 31;
  const int colTiles = dout >> 4;
  const int rowTile  = waveId / colTiles;
  const int colTile  = waveId - rowTile * colTiles;
  const int m0 = rowTile << 4;
  if (m0 >= n) return;                      // wave-uniform: EXEC stays all-1s
  const int n0 = colTile << 4;

  const int ml    = lane & 15;              // m (or n) within tile
  const int khalf = lane >> 4;              // 0 -> K{0,1}, 1 -> K{2,3}
  const float* xrow = X + (size_t)(m0 + ml) * din;
  const int col = n0 + ml;

  v8f accL = {};
  v8f accR = {};
  for (int k0 = 0; k0 < din; k0 += 4) {
    const int k = k0 + (khalf << 1);
    v2f a;  a.x  = xrow[k];              a.y  = xrow[k + 1];
    v2f bL; bL.x = sWl[k * dout + col];  bL.y = sWl[(k + 1) * dout + col];
    v2f bR; bR.x = sWr[k * dout + col];  bR.y = sWr[(k + 1) * dout + col];
    accL = __builtin_amdgcn_wmma_f32_16x16x4_f32(false, a, false, bL,
                                                 (short)0, accL, false, false);
    accR = __builtin_amdgcn_wmma_f32_16x16x4_f32(false, a, false, bR,
                                                 (short)0, accR, false, false);
  }

  const int rowBase = m0 + (khalf << 3);    // lanes 16-31 hold M = 8..15
  const int outCol  = n0 + ml;
#pragma unroll
  for (int i = 0; i < 8; ++i) {
    Y[(size_t)(rowBase + i) * dout + outCol] = accL[i];
    R[(size_t)(rowBase + i) * dout + outCol] = accR[i];
  }
}

// ---------------------------------------------------------------------------
__global__ void fill_zero_f32(float* __restrict__ p, size_t count) {
  size_t i = (size_t)blockIdx.x * blockDim.x + threadIdx.x;
  size_t stride = (size_t)gridDim.x * blockDim.x;
  for (; i < count; i += stride) p[i] = 0.0f;
}

// deg[dst[e]] += 1 for every edge (edge structure is layer-invariant)
__global__ void degree_kernel(const long long* __restrict__ dst,
                              float* __restrict__ deg, long long e_cnt) {
  long long e = (long long)blockIdx.x * blockDim.x + threadIdx.x;
  if (e >= e_cnt) return;
  atomicAdd(&deg[dst[e]], 1.0f);
}

// deginv[i] = 1 / max(deg[i], 1): computed once, multiplied in finalize (x6)
__global__ void recip_kernel(const float* __restrict__ deg,
                             float* __restrict__ deginv, int n) {
  int i = blockIdx.x * blockDim.x + threadIdx.x;
  if (i >= n) return;
  deginv[i] = 1.0f / fmaxf(deg[i], 1.0f);
}

// agg[dst[e], f] += Y[src[e], f]; thread per (edge, feature), dout = 1<<logd.
// Consecutive lanes cover one edge's contiguous feature block -> coalesced
// gather and contiguous atomic targets; Y (12.8 MB) stays L2-resident.
// Group leader prefetches the edge-index stream ~2K edges ahead
// (global_prefetch_b8; speculative, silently dropped if OOB).
__global__ void scatter_add_kernel(const float* __restrict__ Y,
                                   const long long* __restrict__ src,
                                   const long long* __restrict__ dst,
                                   float* __restrict__ agg,
                                   int logd, long long e_cnt, long long total) {
  long long idx = (long long)blockIdx.x * blockDim.x + threadIdx.x;
  if (idx >= total) return;
  const long long e = idx >> logd;
  const int f = (int)(idx & ((1 << logd) - 1));
  if (f == 0) {
    const long long pe = e + 2048;
    if (pe < e_cnt) {
      __builtin_prefetch(&src[pe], 0, 0);
      __builtin_prefetch(&dst[pe], 0, 0);
    }
  }
  const long long s = src[e];
  const long long d = dst[e];
  atomicAdd(&agg[(d << logd) + f], Y[(s << logd) + f]);
}

// out = maybe_relu(agg * deginv + bias + R)
__global__ void finalize_kernel(const float* __restrict__ agg,
                                const float* __restrict__ deginv,
                                const float* __restrict__ R,
                                const float* __restrict__ bias,
                                float* __restrict__ out,
                                int logd, long long total, int doRelu) {
  long long idx = (long long)blockIdx.x * blockDim.x + threadIdx.x;
  if (idx >= total) return;
  const long long i = idx >> logd;
  const int f = (int)(idx & ((1 << logd) - 1));
  float v = fmaf(agg[idx], deginv[i], bias[f] + R[idx]);
  if (doRelu) v = fmaxf(v, 0.0f);
  out[idx] = v;
}

// ---------------------------------------------------------------------------
extern "C" void kernel_launch(void* const* d_in, const int* in_sizes, int n_in,
                              void* d_out, int out_size, void* d_ws, size_t ws_size,
                              hipStream_t stream) {
  (void)in_sizes; (void)n_in; (void)out_size; (void)ws_size;
  const int N = NNODES;
  const long long E = NEDGES;
  static const int DIMS[7] = {128, 32, 32, 32, 32, 32, 16};

  const float*     x   = (const float*)d_in[0];
  const long long* ei  = (const long long*)d_in[1];   // int64 per reference
  const long long* src = ei;
  const long long* dst = ei + E;

  // Workspace layout (floats): h0 | h1 | y | r | agg | deg | deginv (~65 MB)
  float* ws     = (float*)d_ws;
  float* h0     = ws;
  float* h1     = h0  + (size_t)N * 32;
  float* y      = h1  + (size_t)N * 32;
  float* r      = y   + (size_t)N * 32;
  float* agg    = r   + (size_t)N * 32;
  float* deg    = agg + (size_t)N * 32;
  float* deginv = deg + (size_t)N;

  // Degrees + reciprocal: computed once, reused by every layer.
  fill_zero_f32<<<512, 256, 0, stream>>>(deg, (size_t)N);
  degree_kernel<<<(int)((E + 255) / 256), 256, 0, stream>>>(dst, deg, E);
  recip_kernel<<<(N + 255) / 256, 256, 0, stream>>>(deg, deginv, N);

  const float* hin = x;
  for (int l = 0; l < 6; ++l) {
    const int din  = DIMS[l];
    const int dout = DIMS[l + 1];
    const int logd = (dout == 32) ? 5 : 4;
    const float* Wl = (const float*)d_in[2 + 3 * l];
    const float* bl = (const float*)d_in[3 + 3 * l];
    const float* Wr = (const float*)d_in[4 + 3 * l];
    float* hout = (l == 5) ? (float*)d_out : ((l & 1) ? h1 : h0);

    // 1) y = hin @ Wl,  r = hin @ Wr  (WMMA, weights staged in LDS)
    {
      const int rowTiles = N / 16;            // 6250 exactly
      const int colTiles = dout / 16;
      const int waves    = rowTiles * colTiles;
      const int threads  = 256;               // 8 waves / block
      const int blocks   = (waves * 32 + threads - 1) / threads;
      const size_t ldsBytes = (size_t)2 * din * dout * sizeof(float);
      sage_dual_gemm<<<blocks, threads, ldsBytes, stream>>>(hin, Wl, Wr, y, r,
                                                            N, din, dout);
    }

    // 2) agg = 0
    fill_zero_f32<<<2048, 256, 0, stream>>>(agg, (size_t)N * dout);

    // 3) agg[dst] += y[src]   (post-transform: dout floats/edge, L2-resident)
    {
      const long long total = E << logd;
      scatter_add_kernel<<<(int)((total + 255) / 256), 256, 0, stream>>>(
          y, src, dst, agg, logd, E, total);
    }

    // 4) hout = relu?(agg * deginv + bl + r)   (relu after layers 0..4)
    {
      const long long total = (long long)N << logd;
      finalize_kernel<<<(int)((total + 255) / 256), 256, 0, stream>>>(
          agg, deginv, r, bl, hout, logd, total, (l < 5) ? 1 : 0);
    }
    hin = hout;
  }
}